// GRUExtractor_670014898494
// MI455X (gfx1250) — compile-verified
//
#include <hip/hip_runtime.h>
#include <cstdint>

typedef float v2f __attribute__((ext_vector_type(2)));
typedef float v8f __attribute__((ext_vector_type(8)));

namespace {

constexpr int H    = 40;    // hidden size
constexpr int G    = 120;   // 3H stacked gates (r,z,n)
constexpr int I    = 16;    // layer-0 input size
constexpr int T    = 256;   // time steps
constexpr int NT   = 8;     // N tiles of 16 covering 128 (>= 120)
constexpr int KC_I = 4;     // K chunks of 4 over I=16
constexpr int KC_H = 10;    // K chunks of 4 over H=40

// Flat LDS layout (float offsets). Single array so padded-column tile reads
// (cols 120..127 of the last weight array) stay in bounds of the allocation.
constexpr int OFF_WIH0 = 0;                    // 120 x 16
constexpr int OFF_WIH1 = OFF_WIH0 + G * I;     // 120 x 40
constexpr int OFF_WHH1 = OFF_WIH1 + G * H;     // 120 x 40
constexpr int OFF_BRZ0 = OFF_WHH1 + G * H;     // 80: bih0+bhh0 for r|z cols
constexpr int OFF_BNX0 = OFF_BRZ0 + 80;        // 40: bih0 n-gate
constexpr int OFF_BNH0 = OFF_BNX0 + H;         // 40: bhh0 n-gate
constexpr int OFF_BRZ1 = OFF_BNH0 + H;         // 80: bih1+bhh1 for r|z cols
constexpr int OFF_BNX1 = OFF_BRZ1 + 80;        // 40: bih1 n-gate
constexpr int OFF_BNH1 = OFF_BNX1 + H;         // 40: bhh1 n-gate
constexpr int OFF_ACCS = OFF_BNH1 + H;         // 16 x 80: fused r|z pre-activations
constexpr int OFF_AXN  = OFF_ACCS + 16 * 80;   // 16 x 40: x-side n pre-activation
constexpr int OFF_AHN  = OFF_AXN  + 16 * H;    // 16 x 40: h-side n pre-activation
constexpr int OFF_H0   = OFF_AHN  + 16 * H;    // 16 x 40: layer-0 hidden state
constexpr int OFF_H1   = OFF_H0   + 16 * H;    // 16 x 40: layer-1 hidden state
constexpr int SMEM_FLOATS = OFF_H1 + 16 * H + 64;  // tail pad (~63.0 KB total)

__device__ __forceinline__ v8f wmma_f32(v2f a, v2f b, v8f c) {
  // V_WMMA_F32_16X16X4_F32: D(16x16 f32) = A(16x4 f32) * B(4x16 f32) + C
  return __builtin_amdgcn_wmma_f32_16x16x4_f32(
      /*neg_a=*/false, a, /*neg_b=*/false, b,
      /*c_mod=*/(short)0, c, /*reuse_a=*/false, /*reuse_b=*/false);
}

// B fragment for one 4x16 K-chunk: B[k][n] = W[n][k] (W row-major [G][K]).
// Lane l -> n = nt*16 + (l&15), k = kc*4 + 2*(l>>4) + {0,1}.
__device__ __forceinline__ v2f ldsB(const float* w, int K, int nt, int nloc, int koff, int kc) {
  return *(const v2f*)(w + (nt * 16 + nloc) * K + kc * 4 + koff);
}

// C/D 16x16 f32 layout: lane holds fixed col; VGPR v -> row rbase+v.
__device__ __forceinline__ void store_tile(float* dst, int ldw, int col, int rbase, v8f acc) {
  #pragma unroll
  for (int v = 0; v < 8; ++v) dst[(rbase + v) * ldw + col] = acc[v];
}

__device__ __forceinline__ float fsigmoid(float v) { return 1.0f / (1.0f + __expf(-v)); }

}  // namespace

// One wave32 per block; each wave advances 16 batch rows through all T steps of
// both GRU layers, fully fused (hidden state never leaves LDS/VGPRs).
__global__ __launch_bounds__(32) void gru2_fused(
    const float* __restrict__ x,
    const float* __restrict__ Wih0, const float* __restrict__ Whh0,
    const float* __restrict__ bih0, const float* __restrict__ bhh0,
    const float* __restrict__ Wih1, const float* __restrict__ Whh1,
    const float* __restrict__ bih1, const float* __restrict__ bhh1,
    float* __restrict__ out)
{
  __shared__ float smem[SMEM_FLOATS];
  const int lane  = threadIdx.x & 31;
  const int nloc  = lane & 15;            // col-in-tile (B/C/D) and A row
  const int koff  = (lane >> 4) << 1;     // K sub-offset {0, 2}
  const int rbase = (lane < 16) ? 0 : 8;  // C/D row base for this half-wave
  const int bbase = blockIdx.x * 16;

  // ---- stage shared weights/biases to LDS; zero hidden state ----
  for (int i = lane; i < G * I; i += 32) smem[OFF_WIH0 + i] = Wih0[i];
  for (int i = lane; i < G * H; i += 32) smem[OFF_WIH1 + i] = Wih1[i];
  for (int i = lane; i < G * H; i += 32) smem[OFF_WHH1 + i] = Whh1[i];
  for (int i = lane; i < 80; i += 32) {
    smem[OFF_BRZ0 + i] = bih0[i] + bhh0[i];
    smem[OFF_BRZ1 + i] = bih1[i] + bhh1[i];
  }
  for (int i = lane; i < H; i += 32) {
    smem[OFF_BNX0 + i] = bih0[80 + i]; smem[OFF_BNH0 + i] = bhh0[80 + i];
    smem[OFF_BNX1 + i] = bih1[80 + i]; smem[OFF_BNH1 + i] = bhh1[80 + i];
  }
  for (int i = lane; i < 16 * H; i += 32) { smem[OFF_H0 + i] = 0.f; smem[OFF_H1 + i] = 0.f; }

  // ---- pin layer-0 recurrent weight (hottest operand) as B fragments in VGPRs ----
  v2f whh0[KC_H][NT];
  #pragma unroll
  for (int kc = 0; kc < KC_H; ++kc) {
    #pragma unroll
    for (int nt = 0; nt < NT; ++nt) {
      const int n = nt * 16 + nloc;
      v2f w = {0.f, 0.f};
      if (n < G) w = *(const v2f*)(Whh0 + n * H + kc * 4 + koff);
      whh0[kc][nt] = w;
    }
  }

  __syncthreads();
  asm volatile("s_wait_dscnt 0" ::: "memory");

  float* sH0 = smem + OFF_H0;
  float* sH1 = smem + OFF_H1;
  const float* xrow = x + (size_t)(bbase + nloc) * T * I;  // this lane's A-row of x

  #pragma unroll 1
  for (int t = 0; t < T; ++t) {
    // prefetch next step's x fragments (global_prefetch_b8; speculative, OOB-safe)
    if (t + 1 < T) __builtin_prefetch(xrow + (t + 1) * I + koff, 0, 3);

    // =============== layer 0 ===============
    v2f ax[KC_I];
    #pragma unroll
    for (int kc = 0; kc < KC_I; ++kc)
      ax[kc] = *(const v2f*)(xrow + t * I + kc * 4 + koff);
    v2f ah[KC_H];
    #pragma unroll
    for (int kc = 0; kc < KC_H; ++kc)
      ah[kc] = *(const v2f*)(sH0 + nloc * H + kc * 4 + koff);
    asm volatile("s_wait_dscnt 0" ::: "memory");

    #pragma unroll
    for (int nt = 0; nt < NT; ++nt) {
      const int n = nt * 16 + nloc;
      if (nt < 5) {  // r|z columns: chain x-GEMM and h-GEMM into one accumulator
        v8f acc = {};  // inline-0 C operand; biases folded into elementwise phase
        #pragma unroll
        for (int kc = 0; kc < KC_I; ++kc)
          acc = wmma_f32(ax[kc], ldsB(smem + OFF_WIH0, I, nt, nloc, koff, kc), acc);
        #pragma unroll
        for (int kc = 0; kc < KC_H; ++kc)
          acc = wmma_f32(ah[kc], whh0[kc][nt], acc);
        store_tile(smem + OFF_ACCS, 80, n, rbase, acc);
      } else {       // n columns: keep x and h parts separate (r gates the h part)
        v8f accx = {}, acch = {};
        #pragma unroll
        for (int kc = 0; kc < KC_I; ++kc)
          accx = wmma_f32(ax[kc], ldsB(smem + OFF_WIH0, I, nt, nloc, koff, kc), accx);
        #pragma unroll
        for (int kc = 0; kc < KC_H; ++kc)
          acch = wmma_f32(ah[kc], whh0[kc][nt], acch);
        if (n < G) {
          store_tile(smem + OFF_AXN, H, n - 80, rbase, accx);
          store_tile(smem + OFF_AHN, H, n - 80, rbase, acch);
        }
      }
    }
    asm volatile("s_wait_dscnt 0" ::: "memory");

    for (int e = lane; e < 16 * H; e += 32) {  // 20 elements/lane
      const int b = e / H, j = e % H;
      const float rz = smem[OFF_ACCS + b * 80 + j]      + smem[OFF_BRZ0 + j];
      const float zz = smem[OFF_ACCS + b * 80 + 40 + j] + smem[OFF_BRZ0 + 40 + j];
      const float xn = smem[OFF_AXN + b * H + j]        + smem[OFF_BNX0 + j];
      const float hn = smem[OFF_AHN + b * H + j]        + smem[OFF_BNH0 + j];
      const float hp = sH0[b * H + j];
      const float r  = fsigmoid(rz);
      const float z  = fsigmoid(zz);
      const float nn = tanhf(xn + r * hn);
      sH0[b * H + j] = nn + z * (hp - nn);
    }
    asm volatile("s_wait_dscnt 0" ::: "memory");

    // =============== layer 1 (input = fresh h0) ===============
    v2f ax1[KC_H], ah1[KC_H];
    #pragma unroll
    for (int kc = 0; kc < KC_H; ++kc) {
      ax1[kc] = *(const v2f*)(sH0 + nloc * H + kc * 4 + koff);
      ah1[kc] = *(const v2f*)(sH1 + nloc * H + kc * 4 + koff);
    }
    asm volatile("s_wait_dscnt 0" ::: "memory");

    #pragma unroll
    for (int nt = 0; nt < NT; ++nt) {
      const int n = nt * 16 + nloc;
      if (nt < 5) {
        v8f acc = {};
        #pragma unroll
        for (int kc = 0; kc < KC_H; ++kc)
          acc = wmma_f32(ax1[kc], ldsB(smem + OFF_WIH1, H, nt, nloc, koff, kc), acc);
        #pragma unroll
        for (int kc = 0; kc < KC_H; ++kc)
          acc = wmma_f32(ah1[kc], ldsB(smem + OFF_WHH1, H, nt, nloc, koff, kc), acc);
        store_tile(smem + OFF_ACCS, 80, n, rbase, acc);
      } else {
        v8f accx = {}, acch = {};
        #pragma unroll
        for (int kc = 0; kc < KC_H; ++kc)
          accx = wmma_f32(ax1[kc], ldsB(smem + OFF_WIH1, H, nt, nloc, koff, kc), accx);
        #pragma unroll
        for (int kc = 0; kc < KC_H; ++kc)
          acch = wmma_f32(ah1[kc], ldsB(smem + OFF_WHH1, H, nt, nloc, koff, kc), acch);
        if (n < G) {
          store_tile(smem + OFF_AXN, H, n - 80, rbase, accx);
          store_tile(smem + OFF_AHN, H, n - 80, rbase, acch);
        }
      }
    }
    asm volatile("s_wait_dscnt 0" ::: "memory");

    for (int e = lane; e < 16 * H; e += 32) {
      const int b = e / H, j = e % H;
      const float rz = smem[OFF_ACCS + b * 80 + j]      + smem[OFF_BRZ1 + j];
      const float zz = smem[OFF_ACCS + b * 80 + 40 + j] + smem[OFF_BRZ1 + 40 + j];
      const float xn = smem[OFF_AXN + b * H + j]        + smem[OFF_BNX1 + j];
      const float hn = smem[OFF_AHN + b * H + j]        + smem[OFF_BNH1 + j];
      const float hp = sH1[b * H + j];
      const float r  = fsigmoid(rz);
      const float z  = fsigmoid(zz);
      const float nn = tanhf(xn + r * hn);
      sH1[b * H + j] = nn + z * (hp - nn);
    }
    asm volatile("s_wait_dscnt 0" ::: "memory");
  }

  // ---- final layer-1 hidden state -> out (B, H) ----
  for (int e = lane; e < 16 * H; e += 32) {
    const int b = e / H, j = e % H;
    out[(size_t)(bbase + b) * H + j] = sH1[b * H + j];
  }
}

extern "C" void kernel_launch(void* const* d_in, const int* in_sizes, int n_in,
                              void* d_out, int out_size, void* d_ws, size_t ws_size,
                              hipStream_t stream) {
  const float* x    = (const float*)d_in[0];
  const float* Wih0 = (const float*)d_in[1];
  const float* Whh0 = (const float*)d_in[2];
  const float* bih0 = (const float*)d_in[3];
  const float* bhh0 = (const float*)d_in[4];
  const float* Wih1 = (const float*)d_in[5];
  const float* Whh1 = (const float*)d_in[6];
  const float* bih1 = (const float*)d_in[7];
  const float* bhh1 = (const float*)d_in[8];
  float* out = (float*)d_out;

  const int B = in_sizes[0] / (T * I);   // 4096
  const int nblocks = (B + 15) / 16;     // 256 independent wave32 recurrences

  gru2_fused<<<nblocks, 32, 0, stream>>>(x, Wih0, Whh0, bih0, bhh0,
                                         Wih1, Whh1, bih1, bhh1, out);
  (void)n_in; (void)out_size; (void)d_ws; (void)ws_size;
}